// GCN_81535659147394
// MI455X (gfx1250) — compile-verified
//
#include <hip/hip_runtime.h>
#include <hip/hip_bf16.h>

typedef float v2f __attribute__((ext_vector_type(2)));
typedef float v8f __attribute__((ext_vector_type(8)));

// ---------------- degree / normalization ----------------

__global__ void deg_init(float* __restrict__ deg, int n) {
    int i = blockIdx.x * blockDim.x + threadIdx.x;
    if (i < n) deg[i] = 1.0f;  // self-loop contribution
}

__global__ void deg_accum(const long long* __restrict__ dst, float* __restrict__ deg, int e) {
    int i = blockIdx.x * blockDim.x + threadIdx.x;
    if (i < e) atomicAdd(&deg[(int)dst[i]], 1.0f);
}

__global__ void deg_to_dinv(float* __restrict__ deg, int n) {
    int i = blockIdx.x * blockDim.x + threadIdx.x;
    if (i < n) deg[i] = rsqrtf(deg[i]);
}

// ---------------- GEMM: H = (relu?)X @ W, X:[n,64], W:[64,64] ----------------
// One wave computes a 16x64 output tile with V_WMMA_F32_16X16X4_F32.
// A 16x4 f32 frag: lanes 0-15 hold M=lane, K={0,1}; lanes 16-31 hold M=lane-16, K={2,3}.
// B 4x16 f32 frag: lanes 0-15 hold N=lane, K={0,1}; lanes 16-31 hold N=lane-16, K={2,3}.
// D 16x16 f32: VGPR r -> M = r + 8*(lane>=16), N = lane%16.

template <bool RELU>
__global__ void __launch_bounds__(256) gemm64_wmma(const float* __restrict__ X,
                                                   const float* __restrict__ W,
                                                   float* __restrict__ H, int n) {
    const int lane  = threadIdx.x & 31;
    const int wave  = threadIdx.x >> 5;
    const int mt    = blockIdx.x * 8 + wave;   // 16-node tile index
    const int row0  = mt * 16;
    if (row0 >= n) return;

    const int m     = lane & 15;               // M for A-frag, N for B-frag
    const int khalf = (lane >> 4) << 1;        // 0 (lanes 0-15) or 2 (lanes 16-31)

    int arow = row0 + m;
    if (arow >= n) arow = n - 1;               // clamp (loads only; stores are guarded)
    const float* __restrict__ xr = X + (size_t)arow * 64;

    v8f acc0 = {}, acc1 = {}, acc2 = {}, acc3 = {};

#pragma unroll
    for (int kk = 0; kk < 64; kk += 4) {
        v2f a = *(const v2f*)(xr + kk + khalf);
        if (RELU) { a.x = fmaxf(a.x, 0.0f); a.y = fmaxf(a.y, 0.0f); }

        const float* __restrict__ w0 = W + (size_t)(kk + khalf) * 64 + m; // row kk+khalf
        const float* __restrict__ w1 = w0 + 64;                           // row kk+khalf+1

        v2f b0; b0.x = w0[0];  b0.y = w1[0];
        v2f b1; b1.x = w0[16]; b1.y = w1[16];
        v2f b2; b2.x = w0[32]; b2.y = w1[32];
        v2f b3; b3.x = w0[48]; b3.y = w1[48];

        acc0 = __builtin_amdgcn_wmma_f32_16x16x4_f32(false, a, false, b0, (short)0, acc0, false, false);
        acc1 = __builtin_amdgcn_wmma_f32_16x16x4_f32(false, a, false, b1, (short)0, acc1, false, false);
        acc2 = __builtin_amdgcn_wmma_f32_16x16x4_f32(false, a, false, b2, (short)0, acc2, false, false);
        acc3 = __builtin_amdgcn_wmma_f32_16x16x4_f32(false, a, false, b3, (short)0, acc3, false, false);
    }

    const int mh = (lane >> 4) << 3;  // 0 or 8
#pragma unroll
    for (int r = 0; r < 8; ++r) {
        const int orow = row0 + mh + r;
        if (orow < n) {
            float* __restrict__ hp = H + (size_t)orow * 64 + m;
            hp[0]  = acc0[r];
            hp[16] = acc1[r];
            hp[32] = acc2[r];
            hp[48] = acc3[r];
        }
    }
}

// ---------------- out[v] = dinv[v]^2 * h[v] + b  (float4 vectorized) ----------------

__global__ void self_bias(const float* __restrict__ h, const float* __restrict__ dinv,
                          const float* __restrict__ b, float* __restrict__ out, int n) {
    int idx = blockIdx.x * blockDim.x + threadIdx.x;  // over n*16 float4s
    if (idx >= n * 16) return;
    int v = idx >> 4;
    float di = dinv[v];
    float s  = di * di;
    float4 hv = ((const float4*)h)[idx];
    float4 bv = ((const float4*)b)[idx & 15];
    float4 o;
    o.x = s * hv.x + bv.x;
    o.y = s * hv.y + bv.y;
    o.z = s * hv.z + bv.z;
    o.w = s * hv.w + bv.w;
    ((float4*)out)[idx] = o;
}

// ---------------- edge scatter: out[dst] += dinv[src]*dinv[dst]*h[src] ----------------

__global__ void edge_agg(const long long* __restrict__ src, const long long* __restrict__ dst,
                         const float* __restrict__ dinv, const float* __restrict__ h,
                         float* __restrict__ out, int e) {
    int idx = blockIdx.x * blockDim.x + threadIdx.x;
    int eid = idx >> 6;   // 64 threads per edge
    int f   = idx & 63;
    if (eid >= e) return;
    int s = (int)src[eid];
    int d = (int)dst[eid];
    float nrm = dinv[s] * dinv[d];
    atomicAdd(&out[(size_t)d * 64 + f], nrm * h[(size_t)s * 64 + f]);
}

// ---------------- launcher ----------------

extern "C" void kernel_launch(void* const* d_in, const int* in_sizes, int n_in,
                              void* d_out, int out_size, void* d_ws, size_t ws_size,
                              hipStream_t stream) {
    const float*     x    = (const float*)d_in[0];
    const long long* ei   = (const long long*)d_in[1];  // int64 [2, E]
    const float*     W1   = (const float*)d_in[2];
    const float*     b1   = (const float*)d_in[3];
    const float*     W2   = (const float*)d_in[4];
    const float*     b2   = (const float*)d_in[5];
    float*           out  = (float*)d_out;

    const int n = in_sizes[0] / 64;
    const int e = in_sizes[1] / 2;
    const long long* esrc = ei;
    const long long* edst = ei + e;

    float* dinv = (float*)d_ws;               // n floats
    float* bufA = dinv + n;                   // n*64 floats
    float* bufB = bufA + (size_t)n * 64;      // n*64 floats

    const int TB = 256;
    const int nodeBlocks = (n + TB - 1) / TB;
    const int edgeBlocks = (e + TB - 1) / TB;
    const int vecBlocks  = (n * 16 + TB - 1) / TB;
    const int featBlocks = (int)(((long long)e * 64 + TB - 1) / TB);
    const int mtiles     = (n + 15) / 16;
    const int gemmBlocks = (mtiles + 7) / 8;

    // symmetric normalization (shared by both layers)
    deg_init   <<<nodeBlocks, TB, 0, stream>>>(dinv, n);
    deg_accum  <<<edgeBlocks, TB, 0, stream>>>(edst, dinv, e);
    deg_to_dinv<<<nodeBlocks, TB, 0, stream>>>(dinv, n);

    // layer 1: h1 = x @ W1 ; agg1 = D^-1/2 A D^-1/2 h1 + b1
    gemm64_wmma<false><<<gemmBlocks, TB, 0, stream>>>(x, W1, bufA, n);
    self_bias<<<vecBlocks, TB, 0, stream>>>(bufA, dinv, b1, bufB, n);
    edge_agg <<<featBlocks, TB, 0, stream>>>(esrc, edst, dinv, bufA, bufB, e);

    // layer 2: h2 = relu(agg1) @ W2 (relu fused into A-load) ; out = agg + b2
    gemm64_wmma<true><<<gemmBlocks, TB, 0, stream>>>(bufB, W2, bufA, n);
    self_bias<<<vecBlocks, TB, 0, stream>>>(bufA, dinv, b2, out, n);
    edge_agg <<<featBlocks, TB, 0, stream>>>(esrc, edst, dinv, bufA, out, e);
}